// GraphiT_Layer_34823594836037
// MI455X (gfx1250) — compile-verified
//
#include <hip/hip_runtime.h>
#include <hip/hip_bf16.h>

// GraphiT edge-attention, fused for MI455X (gfx1250, wave32).
// B=2, N=256, IN_DIM=512, IN_DIM_E=64, H=8, D=64.
//
// Roofline: dominant work is E = edge_features @ We (8.6 GFLOP fp32).
// Materializing E = ~800MB HBM traffic (~35us @ 23.3TB/s). Fused version
// reads edge_features (33.5MB) exactly once (~1.5us) and is matrix-pipe /
// exp-bound. fp32 WMMA (v_wmma_f32_16x16x4_f32) keeps fp32-exact math.
// ef chunk staging is double-buffered with GLOBAL_LOAD_ASYNC_TO_LDS_B128
// (ASYNCcnt) so the 4KB/chunk copy overlaps the WMMA+exp compute.

typedef __attribute__((ext_vector_type(2))) float v2f;
typedef __attribute__((ext_vector_type(8))) float v8f;

// fp32 WMMA fragment layouts (CDNA5 ISA 7.12.2):
//  A 16x4 : lane<16 -> M=lane,    VGPR0=K0,VGPR1=K1 ; lane>=16 -> M=lane-16, K2/K3
//  B 4x16 : lane<16 -> N=lane,    VGPR0=K0,VGPR1=K1 ; lane>=16 -> N=lane-16, K2/K3
//  C 16x16: VGPR r  : lane<16 -> (M=r, N=lane) ; lane>=16 -> (M=r+8, N=lane-16)
#define WMMA4(a, b, c) \
    __builtin_amdgcn_wmma_f32_16x16x4_f32(false, (a), false, (b), (short)0, (c), false, false)

// Async global->LDS copy, 16B per lane, GV addressing (64-bit VGPR address).
// Tracked by ASYNCcnt; wait with s_wait_asynccnt.
__device__ __forceinline__ void async_g2l_b128(unsigned lds_byte_off, const void* gaddr)
{
    asm volatile("global_load_async_to_lds_b128 %0, %1, off"
                 :
                 : "v"(lds_byte_off), "v"(gaddr)
                 : "memory");
}
__device__ __forceinline__ void wait_async0()
{
    asm volatile("s_wait_asynccnt 0x0" ::: "memory");
}

// ---------------------------------------------------------------------------
// Kernel 1: Q = h@Wq, K = h@Wk * D^-0.5, V = h@Wv.   [512 x 512] @ [512 x 512]
// grid (8, 32, 3), block 128 (4 waves). Wave computes one 16x16 tile.
// ---------------------------------------------------------------------------
__global__ __launch_bounds__(128)
void qkv_proj_kernel(const float* __restrict__ h,
                     const float* __restrict__ Wq,
                     const float* __restrict__ Wk,
                     const float* __restrict__ Wv,
                     float* __restrict__ Qs,
                     float* __restrict__ Ks,
                     float* __restrict__ Vs)
{
    const int lane  = threadIdx.x & 31;
    const int wave  = threadIdx.x >> 5;
    const int mBase = blockIdx.y << 4;
    const int nBase = ((blockIdx.x << 2) + wave) << 4;
    const int which = blockIdx.z;

    const float* W   = (which == 0) ? Wq : (which == 1 ? Wk : Wv);
    float*       Dst = (which == 0) ? Qs : (which == 1 ? Ks : Vs);

    const int ml = lane & 15;       // A row / B,C column within tile
    const int hi = lane >> 4;       // half-wave select
    const int kb = hi << 1;         // K sub-offset (0 or 2)

    const float* aRow = h + (size_t)(mBase + ml) * 512;
    v8f acc = {};
    for (int k = 0; k < 512; k += 4) {
        v2f a = *(const v2f*)(aRow + k + kb);             // contiguous K pair
        v2f b;
        b.x = W[(size_t)(k + kb    ) * 512 + nBase + ml]; // coalesced 16 lanes
        b.y = W[(size_t)(k + kb + 1) * 512 + nBase + ml];
        acc = WMMA4(a, b, acc);
    }
    const float scale = (which == 1) ? 0.125f : 1.0f;     // D^-0.5 = 1/8
#pragma unroll
    for (int r = 0; r < 8; ++r)
        Dst[(size_t)(mBase + r + (hi << 3)) * 512 + nBase + ml] = acc[r] * scale;
}

// ---------------------------------------------------------------------------
// Kernel 2: qk[b,h,i,j] = sum_k Q[b,i,h,k] * Kscaled[b,j,h,k]
// Per (b,h): 256x256x64 GEMM. Both fragments are contiguous float2 loads
// (k is the fastest axis of both operands).
// grid (4, 16, 16), block 128.
// ---------------------------------------------------------------------------
__global__ __launch_bounds__(128)
void qk_scores_kernel(const float* __restrict__ Qs,
                      const float* __restrict__ Ks,
                      float* __restrict__ qk)
{
    const int lane  = threadIdx.x & 31;
    const int wave  = threadIdx.x >> 5;
    const int jBase = ((blockIdx.x << 2) + wave) << 4;
    const int iBase = blockIdx.y << 4;
    const int b     = blockIdx.z >> 3;
    const int hh    = blockIdx.z & 7;

    const int ml = lane & 15, hi = lane >> 4, kb = hi << 1;

    const float* qRow = Qs + (size_t)(b * 256 + iBase + ml) * 512 + hh * 64;
    const float* kRow = Ks + (size_t)(b * 256 + jBase + ml) * 512 + hh * 64;

    v8f acc = {};
#pragma unroll
    for (int k = 0; k < 64; k += 4) {
        v2f a  = *(const v2f*)(qRow + k + kb);
        v2f bf = *(const v2f*)(kRow + k + kb);
        acc = WMMA4(a, bf, acc);
    }
    float* dst = qk + ((size_t)(b * 8 + hh) * 256 + iBase) * 256 + jBase;
#pragma unroll
    for (int r = 0; r < 8; ++r)
        dst[(size_t)(r + (hi << 3)) * 256 + ml] = acc[r];
}

// ---------------------------------------------------------------------------
// Kernel 3: fused edge-attention. One block per (h, i, b); 4 waves cover the
// 64 head columns (4 x 16-col tiles). Loop j in chunks of 16:
//   S[jl, k] = (ef[b,i,j,:] @ We[:, h*64+k]) + qk[b,h,i,j]      (fp32 WMMA)
//   p = exp(clip(S, -5, 5));  f = mask_i*mask_j
//   den += p*f ;  num += p*f*f * V[b,j,h,k]
// out = num / max(den, 1e-6)
//
// LDS (stride 68 floats = 272B rows: bank-conflict-free b64 fragment reads
// AND 16B-aligned async B128 destinations):
//   sWeT : We head-slice TRANSPOSED -> B fragment = one b64 ds_load
//   sEf  : double-buffered ef chunk, filled by global_load_async_to_lds_b128
//          issued one chunk ahead (overlaps WMMA+exp compute).
// grid (8, 256, 2), block 128.
// ---------------------------------------------------------------------------
#define EF_STRIDE 68
#define EF_BUF    (16 * EF_STRIDE)          // floats per buffer (4352 B)

__global__ __launch_bounds__(128)
void fused_edge_attn_kernel(const float* __restrict__ ef,
                            const float* __restrict__ mask,
                            const float* __restrict__ We,
                            const float* __restrict__ Vs,
                            const float* __restrict__ qk,
                            float* __restrict__ out)
{
    alignas(16) __shared__ float sWeT[64 * EF_STRIDE];  // [k_col][K_row]
    alignas(16) __shared__ float sEf[2 * EF_BUF];       // [buf][j_local][K]

    const int tid  = threadIdx.x;
    const int lane = tid & 31;
    const int wave = tid >> 5;
    const int hh   = blockIdx.x;
    const int i    = blockIdx.y;
    const int b    = blockIdx.z;

    const int ml   = lane & 15, hi = lane >> 4, kb = hi << 1;
    const int colB = wave << 4;       // head-local column base for this wave

    // Stage We[:, hh*64 : hh*64+64] transposed into LDS (17 KB).
    for (int idx = tid; idx < 64 * 64; idx += 128) {
        const int K = idx >> 6, kk = idx & 63;
        sWeT[kk * EF_STRIDE + K] = We[(size_t)K * 512 + hh * 64 + kk];
    }

    const float  mi     = mask[b * 256 + i];
    const float* efBase = ef + (size_t)(b * 256 + i) * 256 * 64;  // rows j, 64 each
    const float* mRow   = mask + b * 256;
    const float* qkRow  = qk + ((size_t)(b * 8 + hh) * 256 + i) * 256;
    const float* vBase  = Vs + (size_t)b * 256 * 512 + hh * 64 + colB + ml;

    // LDS byte offset of sEf (low 32 bits of the shared-aperture flat address).
    const unsigned sEfOff = (unsigned)(uintptr_t)(&sEf[0]);

    // Issue async copy of one 16x64 ef chunk into buffer `buf`.
    // 256 x 16B chunks, 2 per thread; chunk c -> row c>>4, cols (c&15)*4.
    auto issueChunk = [&](int j0, int buf) {
        const float* g = efBase + (size_t)j0 * 64;
#pragma unroll
        for (int t = 0; t < 2; ++t) {
            const int c  = tid + t * 128;
            const int jl = c >> 4;
            const int cf = (c & 15) << 2;
            async_g2l_b128(sEfOff + (unsigned)(buf * EF_BUF + jl * EF_STRIDE + cf) * 4u,
                           g + c * 4);
        }
    };

    issueChunk(0, 0);                 // prologue: chunk 0 in flight

    float num = 0.f, den = 0.f;

    for (int j0 = 0; j0 < 256; j0 += 16) {
        const int buf = (j0 >> 4) & 1;

        wait_async0();                // this buffer's B128s have landed in LDS
        __syncthreads();              // ...for every wave (also covers sWeT once)

        if (j0 + 16 < 256)            // overlap next chunk's copy with compute
            issueChunk(j0 + 16, buf ^ 1);

        // 16x64-K fp32 WMMA chain -> S tile (16 j x 16 cols) per wave.
        const float* sA = sEf + buf * EF_BUF;
        v8f c = {};
#pragma unroll
        for (int kk = 0; kk < 64; kk += 4) {
            v2f a  = *(const v2f*)(&sA[ml * EF_STRIDE + kk + kb]);            // b64 ds_load
            v2f bf = *(const v2f*)(&sWeT[(colB + ml) * EF_STRIDE + kk + kb]); // b64 ds_load
            c = WMMA4(a, bf, c);
        }

        // Epilogue: logit add, clip, exp, masked accumulate over 8 rows/lane.
        // qk/mask reads are uniform over the 16 lanes sharing a row: broadcast.
#pragma unroll
        for (int r = 0; r < 8; ++r) {
            const int   jl = r + (hi << 3);
            float s  = c[r] + qkRow[j0 + jl];
            s        = fminf(fmaxf(s, -5.f), 5.f);
            const float p  = __expf(s);
            const float f  = mi * mRow[j0 + jl];
            const float pf = p * f;
            den += pf;                                      // mask applied once
            num += pf * f * vBase[(size_t)(j0 + jl) * 512]; // mask applied twice
        }
    }

    // Lane l and l^16 hold the same column (different j rows): combine halves.
    num += __shfl_xor(num, 16, 32);
    den += __shfl_xor(den, 16, 32);
    if (lane < 16)
        out[(size_t)(b * 256 + i) * 512 + hh * 64 + colB + lane] =
            num / fmaxf(den, 1e-6f);
}

// ---------------------------------------------------------------------------
extern "C" void kernel_launch(void* const* d_in, const int* in_sizes, int n_in,
                              void* d_out, int out_size, void* d_ws, size_t ws_size,
                              hipStream_t stream)
{
    const float* h    = (const float*)d_in[0];   // [2,256,512]
    const float* ef   = (const float*)d_in[1];   // [2,256,256,64]
    const float* mask = (const float*)d_in[2];   // [2,256]
    const float* Wq   = (const float*)d_in[3];   // [512,512]
    const float* Wk   = (const float*)d_in[4];   // [512,512]
    const float* We   = (const float*)d_in[5];   // [64,512]
    const float* Wv   = (const float*)d_in[6];   // [512,512]
    float* out = (float*)d_out;                  // [2,256,512]

    // Workspace layout (floats): Q | Kscaled | V | qk  = 7 MB total.
    float* ws = (float*)d_ws;
    float* Qs = ws;                 // 262144
    float* Ks = ws + 262144;        // 262144
    float* Vs = ws + 524288;        // 262144
    float* qk = ws + 786432;        // 1048576

    qkv_proj_kernel<<<dim3(8, 32, 3), 128, 0, stream>>>(h, Wq, Wk, Wv, Qs, Ks, Vs);
    qk_scores_kernel<<<dim3(4, 16, 16), 128, 0, stream>>>(Qs, Ks, qk);
    fused_edge_attn_kernel<<<dim3(8, 256, 2), 128, 0, stream>>>(ef, mask, We, Vs, qk, out);
}